// EMD1DLoss_47476568490545
// MI455X (gfx1250) — compile-verified
//
#include <hip/hip_runtime.h>
#include <hip/hip_bf16.h>

// EMD1DLoss collapses analytically:
//   loss = sqrt( m*sum(a^2) - 2*sum(a)*sum(b) + n*sum(b^2) ) / n
// -> pure reduction over 2 x 8192 fp32 (64 KB). One workgroup, WMMA-f32
// reduction via the ones-matrix trick (D += Ones(16x4) x B(4x16) sums every
// element of B, independent of B's internal lane layout).

typedef float v2f __attribute__((ext_vector_type(2)));
typedef float v8f __attribute__((ext_vector_type(8)));

#define BLOCK 256
#define WAVES (BLOCK / 32)

__global__ __launch_bounds__(BLOCK) void emd1d_wmma_reduce(
    const float* __restrict__ a, const float* __restrict__ b,
    float* __restrict__ out, int n, int m) {
  const int tid  = threadIdx.x;
  const int lane = tid & 31;
  const int wave = tid >> 5;

  const v2f onesA = {1.0f, 1.0f};  // A = 16x4 all-ones (constant per lane)

  v8f acc_sa  = {};   // accumulates column-sums of a
  v8f acc_sa2 = {};   // accumulates column-sums of a^2
  v8f acc_sb  = {};
  v8f acc_sb2 = {};

  const int nchunks_a = n >> 6;  // 64 elements per chunk
  const int nchunks_b = m >> 6;

  // --- main loops: wave-uniform trip counts -> EXEC all-ones around WMMA ---
  for (int c = wave; c < nchunks_a; c += WAVES) {
    const float* p = a + (c << 6);
    float x0 = p[lane];
    float x1 = p[lane + 32];
    v2f Bv  = {x0, x1};
    v2f Bsq = {x0 * x0, x1 * x1};
    acc_sa  = __builtin_amdgcn_wmma_f32_16x16x4_f32(
        false, onesA, false, Bv,  (short)0, acc_sa,  false, false);
    acc_sa2 = __builtin_amdgcn_wmma_f32_16x16x4_f32(
        false, onesA, false, Bsq, (short)0, acc_sa2, false, false);
  }
  for (int c = wave; c < nchunks_b; c += WAVES) {
    const float* p = b + (c << 6);
    float x0 = p[lane];
    float x1 = p[lane + 32];
    v2f Bv  = {x0, x1};
    v2f Bsq = {x0 * x0, x1 * x1};
    acc_sb  = __builtin_amdgcn_wmma_f32_16x16x4_f32(
        false, onesA, false, Bv,  (short)0, acc_sb,  false, false);
    acc_sb2 = __builtin_amdgcn_wmma_f32_16x16x4_f32(
        false, onesA, false, Bsq, (short)0, acc_sb2, false, false);
  }

  // D rows are identical (A is all-ones); VGPR0 of lanes 0..15 holds the 16
  // column partial sums -> mask upper half to avoid double counting.
  float sa  = (lane < 16) ? acc_sa[0]  : 0.0f;
  float sa2 = (lane < 16) ? acc_sa2[0] : 0.0f;
  float sb  = (lane < 16) ? acc_sb[0]  : 0.0f;
  float sb2 = (lane < 16) ? acc_sb2[0] : 0.0f;

  // scalar tail for n % 64 (dead for n = 8192, kept for generality)
  for (int i = (nchunks_a << 6) + tid; i < n; i += BLOCK) {
    float x = a[i]; sa += x; sa2 += x * x;
  }
  for (int i = (nchunks_b << 6) + tid; i < m; i += BLOCK) {
    float x = b[i]; sb += x; sb2 += x * x;
  }

  __shared__ float s0[BLOCK], s1[BLOCK], s2[BLOCK], s3[BLOCK];
  s0[tid] = sa; s1[tid] = sa2; s2[tid] = sb; s3[tid] = sb2;
  __syncthreads();

  if (tid == 0) {
    float Sa = 0.f, Sa2 = 0.f, Sb = 0.f, Sb2 = 0.f;
    for (int i = 0; i < BLOCK; ++i) {
      Sa += s0[i]; Sa2 += s1[i]; Sb += s2[i]; Sb2 += s3[i];
    }
    float total = (float)m * Sa2 - 2.0f * Sa * Sb + (float)n * Sb2;
    total = total > 0.0f ? total : 0.0f;  // guard fp rounding below zero
    out[0] = sqrtf(total) / (float)n;
  }
}

extern "C" void kernel_launch(void* const* d_in, const int* in_sizes, int n_in,
                              void* d_out, int out_size, void* d_ws, size_t ws_size,
                              hipStream_t stream) {
  (void)n_in; (void)out_size; (void)d_ws; (void)ws_size;
  const float* z_s = (const float*)d_in[0];
  const float* z_t = (const float*)d_in[1];
  float* out = (float*)d_out;
  const int n = in_sizes[0];
  const int m = in_sizes[1];
  emd1d_wmma_reduce<<<1, BLOCK, 0, stream>>>(z_s, z_t, out, n, m);
}